// Model_68066641707582
// MI455X (gfx1250) — compile-verified
//
#include <hip/hip_runtime.h>
#include <hip/hip_bf16.h>

// GraphSAGE (2x SAGEConv mean + edge MLP) for MI455X / gfx1250, fp32 end-to-end.
// GEMMs use V_WMMA_F32_16X16X4_F32 (fp32 in, fp32 acc) since precision must
// match the fp32 reference and the GEMMs are memory-, not compute-, limited.

typedef __attribute__((ext_vector_type(2))) float v2f;
typedef __attribute__((ext_vector_type(8))) float v8f;

#define DD 128
#define NN 50000
#define EE 600000

// ---------------- zero fill ----------------
__global__ void zero_f32(float* __restrict__ p, long n) {
    long i = (long)blockIdx.x * blockDim.x + threadIdx.x;
    long stride = (long)gridDim.x * blockDim.x;
    for (; i < n; i += stride) p[i] = 0.0f;
}

// ------------- scatter-add: one wave32 per edge, float4 per lane -------------
__global__ void scatter_add(const float* __restrict__ feat,
                            const int* __restrict__ src,
                            const int* __restrict__ dst,
                            float* __restrict__ agg,
                            float* __restrict__ deg,   // may be null
                            int nEdges) {
    int wave = threadIdx.x >> 5;
    int lane = threadIdx.x & 31;
    int e = blockIdx.x * 8 + wave;
    if (e >= nEdges) return;
    int s = src[e];
    int d = dst[e];
    const float4 v = *(const float4*)(feat + (long)s * DD + lane * 4);
    float* ap = agg + (long)d * DD + lane * 4;
    unsafeAtomicAdd(ap + 0, v.x);
    unsafeAtomicAdd(ap + 1, v.y);
    unsafeAtomicAdd(ap + 2, v.z);
    unsafeAtomicAdd(ap + 3, v.w);
    if (deg != nullptr && lane == 0) unsafeAtomicAdd(deg + d, 1.0f);
}

// ------------- deg -> 1/max(deg,1) -------------
__global__ void inv_deg(float* __restrict__ deg, int n) {
    int i = blockIdx.x * blockDim.x + threadIdx.x;
    if (i < n) deg[i] = 1.0f / fmaxf(deg[i], 1.0f);
}

// ------------- fused SAGE layer GEMM: out = act(A1@Ws + (A2*invdeg)@Wn + b) -
// One wave per 16x16 C tile. K = 128 per operand, stepped by 4 with
// v_wmma_f32_16x16x4_f32. 50000 rows = 3125 full 16-row tiles (exact).
__global__ void sage_gemm(const float* __restrict__ A1,
                          const float* __restrict__ A2,
                          const float* __restrict__ invdeg,
                          const float* __restrict__ Wself,
                          const float* __restrict__ Wneigh,
                          const float* __restrict__ bias,
                          float* __restrict__ out,
                          int do_relu) {
    const int wave = threadIdx.x >> 5;
    const int lane = threadIdx.x & 31;
    const int tile = blockIdx.x * 8 + wave;   // 3125 blocks * 8 waves = 25000 tiles
    const int nt = tile & 7;                  // 8 column tiles (128/16)
    const int mt = tile >> 3;                 // 3125 row tiles

    const int row = mt * 16 + (lane & 15);    // A-frag row for this lane
    const int col = nt * 16 + (lane & 15);    // B-frag / C column for this lane
    const int khalf = (lane >> 4) << 1;       // lanes 16-31 hold K+2,K+3

    v8f c = {};

    // ---- A1 @ Wself ----
    {
        const float* ap = A1 + (long)row * DD + khalf;
        const float* wp = Wself + (long)khalf * DD + col;
#pragma unroll 8
        for (int k = 0; k < DD; k += 4) {
            v2f a = *(const v2f*)(ap + k);            // K = k+khalf, k+khalf+1
            v2f b;
            b.x = wp[(long)k * DD];                   // W[(k+khalf)  ][col]
            b.y = wp[(long)(k + 1) * DD];             // W[(k+khalf+1)][col]
            c = __builtin_amdgcn_wmma_f32_16x16x4_f32(
                false, a, false, b, (short)0, c, false, false);
        }
    }

    // ---- (A2 * invdeg) @ Wneigh ---- (mean-agg division fused into A load)
    {
        const float idg = invdeg[row];
        const float* ap = A2 + (long)row * DD + khalf;
        const float* wp = Wneigh + (long)khalf * DD + col;
#pragma unroll 8
        for (int k = 0; k < DD; k += 4) {
            v2f a = *(const v2f*)(ap + k);
            a.x *= idg;
            a.y *= idg;
            v2f b;
            b.x = wp[(long)k * DD];
            b.y = wp[(long)(k + 1) * DD];
            c = __builtin_amdgcn_wmma_f32_16x16x4_f32(
                false, a, false, b, (short)0, c, false, false);
        }
    }

    // ---- bias (+ReLU) + store. C/D layout: VGPR i -> row i (lanes 0-15) or
    // row 8+i (lanes 16-31), column = lane&15 within tile.
    const float bv = bias[col];
    const int rbase = mt * 16 + ((lane >> 4) << 3);
#pragma unroll
    for (int i = 0; i < 8; ++i) {
        float v = c[i] + bv;
        if (do_relu) v = fmaxf(v, 0.0f);
        out[(long)(rbase + i) * DD + col] = v;
    }
}

// ------------- edge predictor: score[e] = [h[src],h[dst]] @ Wp + bp ----------
// One wave32 per edge; 4 features per lane per endpoint; wave reduction.
__global__ void edge_score(const float* __restrict__ h2,
                           const int* __restrict__ src,
                           const int* __restrict__ dst,
                           const float* __restrict__ Wp,   // [256,3] row-major
                           const float* __restrict__ bp,   // [3]
                           float* __restrict__ out,        // [E,3]
                           int nEdges) {
    int wave = threadIdx.x >> 5;
    int lane = threadIdx.x & 31;
    int e = blockIdx.x * 8 + wave;
    if (e >= nEdges) return;
    int s = src[e];
    int d = dst[e];
    const float4 a = *(const float4*)(h2 + (long)s * DD + lane * 4);
    const float4 b = *(const float4*)(h2 + (long)d * DD + lane * 4);
    const float av[4] = {a.x, a.y, a.z, a.w};
    const float bv[4] = {b.x, b.y, b.z, b.w};
    float acc0 = 0.f, acc1 = 0.f, acc2 = 0.f;
    const int r = lane * 4;
#pragma unroll
    for (int k = 0; k < 4; ++k) {
        const float* wa = Wp + (long)(r + k) * 3;
        const float* wb = Wp + (long)(128 + r + k) * 3;
        acc0 = fmaf(av[k], wa[0], acc0);
        acc1 = fmaf(av[k], wa[1], acc1);
        acc2 = fmaf(av[k], wa[2], acc2);
        acc0 = fmaf(bv[k], wb[0], acc0);
        acc1 = fmaf(bv[k], wb[1], acc1);
        acc2 = fmaf(bv[k], wb[2], acc2);
    }
#pragma unroll
    for (int off = 16; off > 0; off >>= 1) {
        acc0 += __shfl_xor(acc0, off, 32);
        acc1 += __shfl_xor(acc1, off, 32);
        acc2 += __shfl_xor(acc2, off, 32);
    }
    if (lane == 0) {
        out[(long)e * 3 + 0] = acc0 + bp[0];
        out[(long)e * 3 + 1] = acc1 + bp[1];
        out[(long)e * 3 + 2] = acc2 + bp[2];
    }
}

extern "C" void kernel_launch(void* const* d_in, const int* in_sizes, int n_in,
                              void* d_out, int out_size, void* d_ws, size_t ws_size,
                              hipStream_t stream) {
    const float* x   = (const float*)d_in[0];
    // d_in[1]: e [E,D] — unused by the reference score path
    const int* src   = (const int*)d_in[2];
    const int* dst   = (const int*)d_in[3];
    const float* W1s = (const float*)d_in[4];
    const float* W1n = (const float*)d_in[5];
    const float* b1  = (const float*)d_in[6];
    const float* W2s = (const float*)d_in[7];
    const float* W2n = (const float*)d_in[8];
    const float* b2  = (const float*)d_in[9];
    const float* Wp  = (const float*)d_in[10];
    const float* bp  = (const float*)d_in[11];
    float* out = (float*)d_out;

    // workspace layout (floats): deg[N] | pad | agg[N*128] | h1[N*128] | h2[N*128]
    float* ws  = (float*)d_ws;
    float* deg = ws;
    float* agg = ws + 50176;                 // 16B-aligned offset past deg[N]
    float* h1  = agg + (long)NN * DD;
    float* h2  = h1 + (long)NN * DD;

    const int edgeBlocks = (EE + 7) / 8;     // 8 edges (waves) per 256-thd block
    const int gemmBlocks = (NN / 16) * 8 / 8; // 3125 blocks, 8 tiles each

    // Layer 1: mean agg of x, then relu(x@W1s + agg@W1n + b1)
    zero_f32<<<2048, 256, 0, stream>>>(ws, 50176 + (long)NN * DD);
    scatter_add<<<edgeBlocks, 256, 0, stream>>>(x, src, dst, agg, deg, EE);
    inv_deg<<<(NN + 255) / 256, 256, 0, stream>>>(deg, NN);
    sage_gemm<<<gemmBlocks, 256, 0, stream>>>(x, agg, deg, W1s, W1n, b1, h1, 1);

    // Layer 2: mean agg of h1, then h1@W2s + agg@W2n + b2
    zero_f32<<<2048, 256, 0, stream>>>(agg, (long)NN * DD);
    scatter_add<<<edgeBlocks, 256, 0, stream>>>(h1, src, dst, agg, nullptr, EE);
    sage_gemm<<<gemmBlocks, 256, 0, stream>>>(h1, agg, deg, W2s, W2n, b2, h2, 0);

    // Edge MLP predictor
    edge_score<<<edgeBlocks, 256, 0, stream>>>(h2, src, dst, Wp, bp, out, EE);
}